// AttentionalPropagation_81037442941184
// MI455X (gfx1250) — compile-verified
//
#include <hip/hip_runtime.h>
#include <math.h>

// AttentionalPropagation for MI455X (gfx1250), fp32 end-to-end.
// All matmuls use V_WMMA_F32_16X16X4_F32 (wave32 matrix pipe, full precision).
// Problem is L2/HBM-bound (~45 GFLOP vs ~150MB traffic, 192MB L2), so the
// f32 WMMA path is the right precision/throughput tradeoff.

#define BB 4
#define DD 256
#define NN 2048
#define HH 4
#define HD 64

typedef float v2f __attribute__((ext_vector_type(2)));
typedef float v8f __attribute__((ext_vector_type(8)));

__device__ __forceinline__ v8f wmma_f32(v2f a, v2f b, v8f c) {
  // 8-arg form: (neg_a, A, neg_b, B, c_mod, C, reuse_a, reuse_b)
  return __builtin_amdgcn_wmma_f32_16x16x4_f32(false, a, false, b, (short)0, c,
                                               false, false);
}

// ---------------------------------------------------------------------------
// Generic GEMM: out[b,m,n] = act( sum_k W[m,k] * in[k,n] + bias[m] )
// "in" rows [0,split) come from in1, rows [split,Cin) from in2 (virtual concat).
// Each wave: 16 (M) x 64 (N) output strip, K accumulated 4 at a time via WMMA.
// grid = (NN/256, Cout/16, B), block = 128 (4 waves).
// ---------------------------------------------------------------------------
__global__ __launch_bounds__(128) void gemm_wmma_kernel(
    const float* __restrict__ W, const float* __restrict__ bias,
    const float* __restrict__ in1, const float* __restrict__ in2, int split,
    int Cout, int Cin, float* __restrict__ out, int relu) {
  const int tid = threadIdx.x;
  const int w = tid >> 5;
  const int lane = tid & 31;
  const int l15 = lane & 15;
  const int hlf = lane >> 4;
  const int b = blockIdx.z;
  const int mBase = blockIdx.y * 16;
  const int nBase = blockIdx.x * 256 + w * 64;
  const int m = mBase + l15;  // A-operand row (same for both lane halves)

  const float* base1 = in1 + (size_t)b * (size_t)split * NN;
  const float* base2 = in2 + (size_t)b * (size_t)(Cin - split) * NN;

  v8f c0 = {}, c1 = {}, c2 = {}, c3 = {};

  for (int k0 = 0; k0 < Cin; k0 += 4) {
    // A fragment: W[m, k0+2*hlf .. +1] (contiguous -> one 8B load)
    v2f a = *reinterpret_cast<const v2f*>(&W[(size_t)m * Cin + k0 + 2 * hlf]);
    const float* src = (k0 < split) ? (base1 + (size_t)k0 * NN)
                                    : (base2 + (size_t)(k0 - split) * NN);
    const float* r0 = src + (size_t)(2 * hlf) * NN;
    const float* r1 = r0 + NN;
    const int n0 = nBase + l15;
    v2f b0, b1, b2, b3;
    b0.x = r0[n0];      b0.y = r1[n0];
    b1.x = r0[n0 + 16]; b1.y = r1[n0 + 16];
    b2.x = r0[n0 + 32]; b2.y = r1[n0 + 32];
    b3.x = r0[n0 + 48]; b3.y = r1[n0 + 48];
    c0 = wmma_f32(a, b0, c0);
    c1 = wmma_f32(a, b1, c1);
    c2 = wmma_f32(a, b2, c2);
    c3 = wmma_f32(a, b3, c3);
  }

  float* ob = out + (size_t)b * (size_t)Cout * NN;
#pragma unroll
  for (int j = 0; j < 8; ++j) {
    const int mr = mBase + j + 8 * hlf;  // C/D layout: VGPR j -> M=j or j+8
    const float bv = bias[mr];
    float v0 = c0[j] + bv, v1 = c1[j] + bv, v2 = c2[j] + bv, v3 = c3[j] + bv;
    if (relu) {
      v0 = fmaxf(v0, 0.f); v1 = fmaxf(v1, 0.f);
      v2 = fmaxf(v2, 0.f); v3 = fmaxf(v3, 0.f);
    }
    float* orow = ob + (size_t)mr * NN + nBase + l15;
    orow[0] = v0; orow[16] = v1; orow[32] = v2; orow[48] = v3;
  }
}

// ---------------------------------------------------------------------------
// Flash attention per (b, h):  S = q^T k / 8, online softmax over m,
// P = exp(S - max)*edge, msg^T[n,d] = P · V^T, normalize by Z (edge-free sum).
// Channel layout from torch-style .view: channel(hd, h) = hd*H + h.
// Block = 128 (4 waves): K/V 64x16 tiles staged in LDS shared by the 4 waves,
// each wave owns a 16-row n-tile. grid = (NN/64, H, B).
// ---------------------------------------------------------------------------
__global__ __launch_bounds__(128) void attn_kernel(
    const float* __restrict__ q, const float* __restrict__ k,
    const float* __restrict__ v, const float* __restrict__ edge,
    float* __restrict__ msg) {
  __shared__ float lds_k[HD * 16];
  __shared__ float lds_v[HD * 16];
  __shared__ float sbuf[4][16 * 16];
  __shared__ float fbuf[4][16];

  const int tid = threadIdx.x;
  const int w = tid >> 5;
  const int lane = tid & 31;
  const int l15 = lane & 15;
  const int hlf = lane >> 4;
  const int b = blockIdx.z, h = blockIdx.y;
  const int nTile = blockIdx.x * 64 + w * 16;

  const size_t HN = (size_t)HH * NN;  // stride between consecutive hd channels
  const float* qb = q + ((size_t)b * DD + h) * NN;
  const float* kb = k + ((size_t)b * DD + h) * NN;
  const float* vb = v + ((size_t)b * DD + h) * NN;

  // Preload all 16 q^T A-fragments (M=n rows, K=hd in chunks of 4).
  v2f a[16];
  {
    const int nRow = nTile + l15;
#pragma unroll
    for (int t = 0; t < 16; ++t) {
      const int hd = 4 * t + 2 * hlf;
      a[t].x = qb[(size_t)hd * HN + nRow];
      a[t].y = qb[(size_t)(hd + 1) * HN + nRow];
    }
  }

  v8f acc0 = {}, acc1 = {}, acc2 = {}, acc3 = {};
  float rowmax = -INFINITY, rowsum = 0.f;

  // Cooperative K/V tile staging: thread -> (hd = tid/2, m half = (tid&1)*8)
  const int ch = tid >> 1;
  const int mh = (tid & 1) * 8;
  const float* kg0 = kb + (size_t)ch * HN + mh;
  const float* vg0 = vb + (size_t)ch * HN + mh;
  float* lk = &lds_k[ch * 16 + mh];
  float* lv = &lds_v[ch * 16 + mh];

  float* sb = sbuf[w];
  float* fb = fbuf[w];
  const float* eg0 =
      edge + ((size_t)b * NN + (size_t)(nTile + l15)) * (size_t)NN + 8 * hlf;

  for (int mt = 0; mt < NN / 16; ++mt) {
    const int mBase = mt * 16;
    {
      const float4* kg = reinterpret_cast<const float4*>(kg0 + mBase);
      const float4* vg = reinterpret_cast<const float4*>(vg0 + mBase);
      float4 ka = kg[0], kbv = kg[1];
      float4 va = vg[0], vbv = vg[1];
      reinterpret_cast<float4*>(lk)[0] = ka;
      reinterpret_cast<float4*>(lk)[1] = kbv;
      reinterpret_cast<float4*>(lv)[0] = va;
      reinterpret_cast<float4*>(lv)[1] = vbv;
    }
    __syncthreads();

    // S tile: 16 WMMAs over K=64
    v8f c = {};
#pragma unroll
    for (int t = 0; t < 16; ++t) {
      const int hd = 4 * t + 2 * hlf;
      v2f bk;
      bk.x = lds_k[hd * 16 + l15];
      bk.y = lds_k[(hd + 1) * 16 + l15];
      c = wmma_f32(a[t], bk, c);
    }
#pragma unroll
    for (int j = 0; j < 8; ++j) c[j] *= 0.125f;  // 1/sqrt(64)

    // Spill S tile to per-wave LDS in [n][m] layout for transpose + softmax.
#pragma unroll
    for (int j = 0; j < 8; ++j) sb[(j + 8 * hlf) * 16 + l15] = c[j];
    __syncthreads();

    // Online softmax: lane handles row n=l15, columns 8*hlf..8*hlf+7.
    float4 s0 = *reinterpret_cast<const float4*>(&sb[l15 * 16 + 8 * hlf]);
    float4 s1 = *reinterpret_cast<const float4*>(&sb[l15 * 16 + 8 * hlf + 4]);
    float tmax = fmaxf(fmaxf(fmaxf(s0.x, s0.y), fmaxf(s0.z, s0.w)),
                       fmaxf(fmaxf(s1.x, s1.y), fmaxf(s1.z, s1.w)));
    tmax = fmaxf(tmax, __shfl_xor(tmax, 16, 32));
    const float newmax = fmaxf(rowmax, tmax);
    const float fscale = __expf(rowmax - newmax);
    rowmax = newmax;

    const float4* eg = reinterpret_cast<const float4*>(eg0 + mBase);
    float4 e0 = eg[0], e1 = eg[1];
    float p0 = __expf(s0.x - newmax), p1 = __expf(s0.y - newmax);
    float p2 = __expf(s0.z - newmax), p3 = __expf(s0.w - newmax);
    float p4 = __expf(s1.x - newmax), p5 = __expf(s1.y - newmax);
    float p6 = __expf(s1.z - newmax), p7 = __expf(s1.w - newmax);
    float psum = ((p0 + p1) + (p2 + p3)) + ((p4 + p5) + (p6 + p7));
    float4 pe0 = make_float4(p0 * e0.x, p1 * e0.y, p2 * e0.z, p3 * e0.w);
    float4 pe1 = make_float4(p4 * e1.x, p5 * e1.y, p6 * e1.z, p7 * e1.w);
    *reinterpret_cast<float4*>(&sb[l15 * 16 + 8 * hlf]) = pe0;
    *reinterpret_cast<float4*>(&sb[l15 * 16 + 8 * hlf + 4]) = pe1;
    psum += __shfl_xor(psum, 16, 32);
    rowsum = rowsum * fscale + psum;
    if (lane < 16) fb[lane] = fscale;  // per-row rescale factor
    __syncthreads();

    // Rescale accumulators (row of element j is j + 8*hlf, not lane).
#pragma unroll
    for (int j = 0; j < 8; ++j) {
      const float fj = fb[j + 8 * hlf];
      acc0[j] *= fj; acc1[j] *= fj; acc2[j] *= fj; acc3[j] *= fj;
    }

    // msg^T[n, d] += P[n, m] * V^T[m, d]  (16 WMMAs: 4 K-chunks x 4 d-tiles)
#pragma unroll
    for (int t = 0; t < 4; ++t) {
      const int co = 4 * t + 2 * hlf;
      v2f pf;
      pf.x = sb[l15 * 16 + co];
      pf.y = sb[l15 * 16 + co + 1];
      v2f vf;
      vf.x = lds_v[l15 * 16 + co];        vf.y = lds_v[l15 * 16 + co + 1];
      acc0 = wmma_f32(pf, vf, acc0);
      vf.x = lds_v[(16 + l15) * 16 + co]; vf.y = lds_v[(16 + l15) * 16 + co + 1];
      acc1 = wmma_f32(pf, vf, acc1);
      vf.x = lds_v[(32 + l15) * 16 + co]; vf.y = lds_v[(32 + l15) * 16 + co + 1];
      acc2 = wmma_f32(pf, vf, acc2);
      vf.x = lds_v[(48 + l15) * 16 + co]; vf.y = lds_v[(48 + l15) * 16 + co + 1];
      acc3 = wmma_f32(pf, vf, acc3);
    }
    __syncthreads();  // protect lds_k/lds_v reuse next iteration
  }

  // Normalize by Z and scatter msg[b, hd*H+h, n].
  if (lane < 16) fb[lane] = rowsum;
  __syncthreads();
  float* mo = msg + ((size_t)b * DD + h) * NN;
#pragma unroll
  for (int j = 0; j < 8; ++j) {
    const int row = j + 8 * hlf;
    const float inv = 1.0f / fb[row];
    const size_t n = (size_t)(nTile + row);
    mo[(size_t)(l15)*HN + n]      = acc0[j] * inv;
    mo[(size_t)(16 + l15) * HN + n] = acc1[j] * inv;
    mo[(size_t)(32 + l15) * HN + n] = acc2[j] * inv;
    mo[(size_t)(48 + l15) * HN + n] = acc3[j] * inv;
  }
}

// ---------------------------------------------------------------------------
extern "C" void kernel_launch(void* const* d_in, const int* in_sizes, int n_in,
                              void* d_out, int out_size, void* d_ws,
                              size_t ws_size, hipStream_t stream) {
  const float* x      = (const float*)d_in[0];
  const float* source = (const float*)d_in[1];
  const float* edge   = (const float*)d_in[2];
  const float* Wq = (const float*)d_in[3];
  const float* bq = (const float*)d_in[4];
  const float* Wk = (const float*)d_in[5];
  const float* bk = (const float*)d_in[6];
  const float* Wv = (const float*)d_in[7];
  const float* bv = (const float*)d_in[8];
  const float* Wm = (const float*)d_in[9];
  const float* bm = (const float*)d_in[10];
  const float* W1 = (const float*)d_in[11];
  const float* b1 = (const float*)d_in[12];
  const float* W2 = (const float*)d_in[13];
  const float* b2 = (const float*)d_in[14];
  float* out = (float*)d_out;

  float* ws = (float*)d_ws;
  const size_t BDN = (size_t)BB * DD * NN;  // 2,097,152 floats
  float* qb   = ws;
  float* kbuf = qb + BDN;
  float* vbuf = kbuf + BDN;
  float* msgb = vbuf + BDN;
  float* msge = msgb + BDN;
  float* h1b  = msge + BDN;  // occupies 2*BDN

  const dim3 blk(128);
  const dim3 gP(NN / 256, DD / 16, BB);

  gemm_wmma_kernel<<<gP, blk, 0, stream>>>(Wq, bq, x, x, DD, DD, DD, qb, 0);
  gemm_wmma_kernel<<<gP, blk, 0, stream>>>(Wk, bk, source, source, DD, DD, DD,
                                           kbuf, 0);
  gemm_wmma_kernel<<<gP, blk, 0, stream>>>(Wv, bv, source, source, DD, DD, DD,
                                           vbuf, 0);

  const dim3 gA(NN / 64, HH, BB);
  attn_kernel<<<gA, blk, 0, stream>>>(qb, kbuf, vbuf, edge, msgb);

  gemm_wmma_kernel<<<gP, blk, 0, stream>>>(Wm, bm, msgb, msgb, DD, DD, DD,
                                           msge, 0);
  const dim3 g1(NN / 256, (2 * DD) / 16, BB);
  gemm_wmma_kernel<<<g1, blk, 0, stream>>>(W1, b1, x, msge, DD, 2 * DD, 2 * DD,
                                           h1b, 1);
  gemm_wmma_kernel<<<gP, blk, 0, stream>>>(W2, b2, h1b, h1b, 2 * DD, DD,
                                           2 * DD, out, 0);
}